// UCBSampler_90666759618579
// MI455X (gfx1250) — compile-verified
//
#include <hip/hip_runtime.h>
#include <hip/hip_bf16.h>
#include <math.h>
#include <stdint.h>

// UCB argmax over rows: rewards[i,j] = probs[i,j] + 0.5*sqrt(log(i+1)/(1 + i*(1 + i*N + j)))
// out[i] = argmax_j rewards[i,j]   (first occurrence on ties)
//
// Memory-bound streaming reduction: 512 MB in, 16 KB out. One block per row,
// gfx1250 async global->LDS prefetch pipeline (ASYNCcnt) + wave32 shuffle reduce.

#define TPB    256
#define DEPTH  4            // async prefetch pipeline depth (tiles in flight)
#define TILE_F (TPB * 4)    // 1024 floats per tile (16 B per thread per tile)

#if defined(__has_builtin)
#  if __has_builtin(__builtin_amdgcn_global_load_async_to_lds_b128)
#    define HAVE_ASYNC_LDS 1
#  else
#    define HAVE_ASYNC_LDS 0
#  endif
#  if __has_builtin(__builtin_amdgcn_s_wait_asynccnt)
#    define HAVE_WAIT_ASYNC 1
#  else
#    define HAVE_WAIT_ASYNC 0
#  endif
#else
#  define HAVE_ASYNC_LDS 0
#  define HAVE_WAIT_ASYNC 0
#endif

#if HAVE_WAIT_ASYNC
#  define ASYNC_WAIT(n) do { __builtin_amdgcn_s_wait_asynccnt(n); asm volatile("" ::: "memory"); } while (0)
#else
#  define ASYNC_WAIT(n) asm volatile("s_wait_asynccnt " #n ::: "memory")
#endif

#if HAVE_ASYNC_LDS
typedef int v4i __attribute__((ext_vector_type(4)));
typedef __attribute__((address_space(1))) v4i* as1_v4i_ptr;
typedef __attribute__((address_space(3))) v4i* as3_v4i_ptr;

// Per ISA §10.2 the LDS aperture maps addr[31:0] directly to the LDS byte
// address, so truncating the generic pointer yields a valid AS3 pointer.
// Generic pointers to global memory are numerically identical to AS1.
__device__ __forceinline__ void async_copy_b128(const float* gsrc, void* lds_dst) {
  __builtin_amdgcn_global_load_async_to_lds_b128(
      (as1_v4i_ptr)(uintptr_t)gsrc,
      (as3_v4i_ptr)(unsigned int)(uintptr_t)lds_dst,
      /*offset=*/0, /*cpol=*/0);
}
#endif

// One element: denom = 1 + i*(t2 + j)  with t2 = 1 + i*N  (fp32, mirroring the
// reference's op order); bonus = c * rsq(denom) with c = 0.5*sqrt(log(i+1)).
#define UCB_UPD(pval, jidx)                                              \
  do {                                                                   \
    float _dn = fmaf(fi, t2 + (float)(jidx), 1.0f);                      \
    float _r  = fmaf(c, __builtin_amdgcn_rsqf(_dn), (pval));             \
    if (_r > bestv) { bestv = _r; bestj = (jidx); }                      \
  } while (0)

__global__ __launch_bounds__(TPB) void ucb_argmax_rows(const float* __restrict__ probs,
                                                       float* __restrict__ out,
                                                       int n_nodes) {
  const int row = (int)blockIdx.x;
  const int tid = (int)threadIdx.x;
  const float* __restrict__ rowp = probs + (size_t)row * (size_t)n_nodes;

  const float fi = (float)row;
  const float L  = logf(fi + 1.0f);
  const float c  = 0.5f * sqrtf(L);              // 0 for row 0 -> bonus == 0
  const float t2 = 1.0f + fi * (float)n_nodes;   // fp32 like the reference

  float bestv = -__builtin_inff();
  int   bestj = 0;

  const int NT   = n_nodes / TILE_F;   // dense float4 tiles
  const int coff = tid << 2;           // this thread's float offset within a tile

#if HAVE_ASYNC_LDS
  __shared__ float4 stage[DEPTH * TPB];   // 16 KB, thread-private 16B slots

  // Prologue: fill the pipeline.
  {
    const int npro = (NT < DEPTH) ? NT : DEPTH;
    for (int t = 0; t < npro; ++t)
      async_copy_b128(rowp + (size_t)t * TILE_F + coff, &stage[t * TPB + tid]);
  }
#endif

  for (int t = 0; t < NT; ++t) {
    const int jbase = t * TILE_F + coff;
    float4 v;
#if HAVE_ASYNC_LDS
    {
      // tiles issued so far = min(NT, t+DEPTH); need tile t complete, so the
      // allowed outstanding count is issued-(t+1). Immediates only.
      const int issued = (t + DEPTH < NT) ? (t + DEPTH) : NT;
      const int allow  = issued - (t + 1);
      switch (allow) {
        case 0:  ASYNC_WAIT(0); break;
        case 1:  ASYNC_WAIT(1); break;
        case 2:  ASYNC_WAIT(2); break;
        default: ASYNC_WAIT(3); break;
      }
      v = stage[(t & (DEPTH - 1)) * TPB + tid];   // own slot: no barrier needed
    }
#else
    v = *reinterpret_cast<const float4*>(rowp + jbase);
#endif

    UCB_UPD(v.x, jbase + 0);
    UCB_UPD(v.y, jbase + 1);
    UCB_UPD(v.z, jbase + 2);
    UCB_UPD(v.w, jbase + 3);

#if HAVE_ASYNC_LDS
    // Value already consumed into VGPRs; safe to reuse the slot.
    const int tn = t + DEPTH;
    if (tn < NT)
      async_copy_b128(rowp + (size_t)tn * TILE_F + coff,
                      &stage[(tn & (DEPTH - 1)) * TPB + tid]);
#endif
  }

  // Vector tail (columns [NT*TILE_F, n_nodes) in aligned float4 chunks).
  for (int jt = NT * TILE_F + coff; jt + 3 < n_nodes; jt += TILE_F) {
    float4 v = *reinterpret_cast<const float4*>(rowp + jt);
    UCB_UPD(v.x, jt + 0);
    UCB_UPD(v.y, jt + 1);
    UCB_UPD(v.z, jt + 2);
    UCB_UPD(v.w, jt + 3);
  }
  // Scalar tail if n_nodes % 4 != 0.
  for (int js = (n_nodes & ~3) + tid; js < n_nodes; js += TPB) {
    UCB_UPD(rowp[js], js);
  }

  // Wave32 butterfly reduce: (max value, min index on ties) — matches argmax
  // first-occurrence semantics.
#pragma unroll
  for (int m = 16; m >= 1; m >>= 1) {
    float ov = __shfl_xor(bestv, m, 32);
    int   oj = __shfl_xor(bestj, m, 32);
    if (ov > bestv || (ov == bestv && oj < bestj)) { bestv = ov; bestj = oj; }
  }

  __shared__ float swv[TPB / 32];
  __shared__ int   swj[TPB / 32];
  const int lane = tid & 31;
  const int wv   = tid >> 5;
  if (lane == 0) { swv[wv] = bestv; swj[wv] = bestj; }
  __syncthreads();

  if (tid == 0) {
    float bv = swv[0];
    int   bj = swj[0];
#pragma unroll
    for (int w = 1; w < TPB / 32; ++w) {
      if (swv[w] > bv || (swv[w] == bv && swj[w] < bj)) { bv = swv[w]; bj = swj[w]; }
    }
    out[row] = (float)bj;   // index <= 31999: exact in fp32
  }
}

extern "C" void kernel_launch(void* const* d_in, const int* in_sizes, int n_in,
                              void* d_out, int out_size, void* d_ws, size_t ws_size,
                              hipStream_t stream) {
  (void)n_in; (void)d_ws; (void)ws_size;
  const float* probs = (const float*)d_in[0];
  float* out = (float*)d_out;

  const int rows = out_size;                                    // 4096
  if (rows <= 0) return;
  const int n_nodes = (int)(in_sizes[0] / rows);                // 32000

  ucb_argmax_rows<<<dim3((unsigned)rows), dim3(TPB), 0, stream>>>(probs, out, n_nodes);
}